// AUCDomainAdapation_20031727468649
// MI455X (gfx1250) — compile-verified
//
#include <hip/hip_runtime.h>

#define NCLS  64
#define NSAMP 1024
#define EPSV  0.05f
#define WARM_UP 10

typedef __attribute__((ext_vector_type(2))) float v2f;
typedef __attribute__((ext_vector_type(8))) float v8f;

// softplus(z) = log1p(exp(z)) computed stably as log1p(exp(-|z|)) + max(z,0)
__device__ __forceinline__ float softplus_f(float z) {
    return log1pf(__expf(-fabsf(z))) + fmaxf(z, 0.0f);
}
// log_loss(x) = softplus(-(x-eps)) + softplus(x+eps)
__device__ __forceinline__ float pair_loss(float x) {
    return softplus_f(-(x - EPSV)) + softplus_f(x + EPSV);
}

// ---------------------------------------------------------------------------
// K1: row softmax for all 4 logit matrices; argmax of pta rows -> pseudo label
// One wave32 per row; each lane owns 2 contiguous classes (float2).
// grid = 512 blocks x 256 threads = 4096 waves = 4 matrices x 1024 rows
// ---------------------------------------------------------------------------
__global__ void softmax_kernel(const float* __restrict__ ys,  const float* __restrict__ ysa,
                               const float* __restrict__ yt,  const float* __restrict__ yta,
                               float* __restrict__ ps,  float* __restrict__ psa,
                               float* __restrict__ pt,  float* __restrict__ pta,
                               int* __restrict__ pseudo) {
    const int lane = threadIdx.x & 31;
    const int wave = threadIdx.x >> 5;
    const int gr   = blockIdx.x * 8 + wave;   // 0..4095
    const int mat  = gr >> 10;
    const int row  = gr & 1023;
    const float* src = (mat == 0) ? ys : (mat == 1) ? ysa : (mat == 2) ? yt : yta;
    float*       dst = (mat == 0) ? ps : (mat == 1) ? psa : (mat == 2) ? pt : pta;

    __builtin_prefetch(src + row * NCLS, 0, 3);   // global_prefetch_b8

    const float2 v = *(const float2*)(src + row * NCLS + lane * 2);
    float m = fmaxf(v.x, v.y);
    #pragma unroll
    for (int off = 16; off >= 1; off >>= 1) m = fmaxf(m, __shfl_xor(m, off, 32));
    const float ex = __expf(v.x - m), ey = __expf(v.y - m);
    float s = ex + ey;
    #pragma unroll
    for (int off = 16; off >= 1; off >>= 1) s += __shfl_xor(s, off, 32);
    const float px = ex / s, py = ey / s;
    float2 o; o.x = px; o.y = py;
    *(float2*)(dst + row * NCLS + lane * 2) = o;

    if (mat == 3) {  // argmax with first-index tie-break (matches jnp.argmax)
        float bv; int bi;
        if (py > px) { bv = py; bi = lane * 2 + 1; } else { bv = px; bi = lane * 2; }
        #pragma unroll
        for (int off = 16; off >= 1; off >>= 1) {
            const float ov = __shfl_xor(bv, off, 32);
            const int   oi = __shfl_xor(bi, off, 32);
            if (ov > bv || (ov == bv && oi < bi)) { bv = ov; bi = oi; }
        }
        if (lane == 0) pseudo[row] = bi;
    }
}

// ---------------------------------------------------------------------------
// K2: class histograms (integer LDS atomics -> deterministic) and fac_c
// ---------------------------------------------------------------------------
__global__ void fac_kernel(const int* __restrict__ labels, const int* __restrict__ pseudo,
                           float* __restrict__ facS, float* __restrict__ facT) {
    __shared__ int cs[NCLS], ct[NCLS];
    const int tid = threadIdx.x;  // 256 threads
    if (tid < NCLS) { cs[tid] = 0; ct[tid] = 0; }
    __syncthreads();
    for (int i = tid; i < NSAMP; i += 256) {
        atomicAdd(&cs[labels[i]], 1);
        atomicAdd(&ct[pseudo[i]], 1);
    }
    __syncthreads();
    if (tid < NCLS) {
        int n = cs[tid];
        facS[tid] = (n > 0 && n < NSAMP) ? 1.0f / ((float)n * (float)(NSAMP - n)) : 0.0f;
        n = ct[tid];
        facT[tid] = (n > 0 && n < NSAMP) ? 1.0f / ((float)n * (float)(NSAMP - n)) : 0.0f;
    }
}

// ---------------------------------------------------------------------------
// K3: per-(class, term) masked pair sum.
//   term 0: empirical   arg = 4*(1-(p_i - p_j)),         x = ps[:,c]
//   term 1: src disc    arg = x_i - x_j, x = 2*(psa-ps)[:,c]
//   term 2: tgt disc    arg = x_i - x_j, x = 2*(pta-pt)[:,c]  (pseudo labels)
// Column staged in LDS (4KB). Block reduction finishes with a fp32 WMMA
// (16x16x4 f32, all-ones B) that sums 64 partials in one matrix op.
// ---------------------------------------------------------------------------
__global__ void pairsum_kernel(const float* __restrict__ ps,  const float* __restrict__ psa,
                               const float* __restrict__ pt,  const float* __restrict__ pta,
                               const int* __restrict__ labels, const int* __restrict__ pseudo,
                               const float* __restrict__ facS, const float* __restrict__ facT,
                               const int* __restrict__ epoch,  float* __restrict__ per) {
    const int c    = blockIdx.x;
    const int term = blockIdx.y;
    const int tid  = threadIdx.x;   // 256 threads
    __shared__ float          xsm[NSAMP];
    __shared__ unsigned char  pf[NSAMP];
    __shared__ float          red[256];
    __shared__ float          fold[64];

    if (term == 2 && epoch[0] < WARM_UP) {   // uniform branch: whole block exits
        if (tid == 0) per[term * NCLS + c] = 0.0f;
        return;
    }
    const int* lab = (term == 2) ? pseudo : labels;

    for (int i = tid; i < NSAMP; i += 256) {
        pf[i] = (lab[i] == c);
        float x;
        if (term == 0)      x = ps[i * NCLS + c];
        else if (term == 1) x = 2.0f * (psa[i * NCLS + c] - ps[i * NCLS + c]);
        else                x = 2.0f * (pta[i * NCLS + c] - pt[i * NCLS + c]);
        xsm[i] = x;
    }
    __syncthreads();

    float acc = 0.0f;
    for (int i = 0; i < NSAMP; ++i) {
        if (!pf[i]) continue;                 // only positives of class c
        const float xi = xsm[i];
        if (term == 0) {
            for (int j = tid; j < NSAMP; j += 256) {
                if (pf[j]) continue;          // only negatives of class c
                acc += pair_loss(4.0f * (1.0f - (xi - xsm[j])));
            }
        } else {
            for (int j = tid; j < NSAMP; j += 256) {
                if (pf[j]) continue;
                acc += pair_loss(xi - xsm[j]);
            }
        }
    }

    red[tid] = acc;
    __syncthreads();
    if (tid < 64) fold[tid] = ((red[tid] + red[tid + 64]) + red[tid + 128]) + red[tid + 192];
    __syncthreads();

    if (tid < 32) {                           // wave 0, EXEC all ones
        // Bijective fill of the 16x4 f32 A operand with the 64 partials;
        // B = all-ones => D rows are row-sums; total = sum of D[:,0].
        const int base = (tid < 16) ? (tid * 4) : ((tid - 16) * 4 + 2);
        v2f a; a.x = fold[base]; a.y = fold[base + 1];
        v2f b; b.x = 1.0f; b.y = 1.0f;
        v8f cacc = {};
        cacc = __builtin_amdgcn_wmma_f32_16x16x4_f32(
            /*neg_a=*/false, a, /*neg_b=*/false, b,
            /*c_mod=*/(short)0, cacc, /*reuse_a=*/false, /*reuse_b=*/false);
        float s = ((((((cacc[0] + cacc[1]) + cacc[2]) + cacc[3]) + cacc[4])
                    + cacc[5]) + cacc[6]) + cacc[7];       // lane0: rows 0-7, lane16: rows 8-15
        const float s16 = __shfl(s, 16, 32);
        if (tid == 0) {
            const float fac = (term == 2) ? facT[c] : facS[c];
            per[term * NCLS + c] = fac * (s + s16);
        }
    }
}

// ---------------------------------------------------------------------------
// K4: deterministic fixed-order combine of the 192 per-(term,class) results
// ---------------------------------------------------------------------------
__global__ void combine_kernel(const float* __restrict__ per, float* __restrict__ out) {
    if (threadIdx.x == 0) {
        float e = 0.0f, sd = 0.0f, td = 0.0f;
        for (int c = 0; c < NCLS; ++c) {
            e  += per[c];
            sd += per[NCLS + c];
            td += per[2 * NCLS + c];
        }
        out[0] = 0.25f * e;                       // 0.25 * empirical
        out[1] = -0.5f * sd + 0.25f * td;         // -BETA2*0.5*src + BETA1*0.25*tgt
    }
}

extern "C" void kernel_launch(void* const* d_in, const int* in_sizes, int n_in,
                              void* d_out, int out_size, void* d_ws, size_t ws_size,
                              hipStream_t stream) {
    (void)in_sizes; (void)n_in; (void)out_size; (void)ws_size;
    const float* y_s   = (const float*)d_in[0];
    const float* y_sa  = (const float*)d_in[1];
    const int*   lab   = (const int*)  d_in[2];
    const float* y_t   = (const float*)d_in[3];
    const float* y_ta  = (const float*)d_in[4];
    const int*   epoch = (const int*)  d_in[5];

    float* ws = (float*)d_ws;
    const int NC = NSAMP * NCLS;
    float* ps     = ws;
    float* psa    = ws + NC;
    float* pt     = ws + 2 * NC;
    float* pta    = ws + 3 * NC;
    int*   pseudo = (int*)(ws + 4 * NC);
    float* facS   = ws + 4 * NC + NSAMP;
    float* facT   = facS + NCLS;
    float* per    = facT + NCLS;
    float* out    = (float*)d_out;

    softmax_kernel<<<512, 256, 0, stream>>>(y_s, y_sa, y_t, y_ta, ps, psa, pt, pta, pseudo);
    fac_kernel<<<1, 256, 0, stream>>>(lab, pseudo, facS, facT);
    pairsum_kernel<<<dim3(NCLS, 3), 256, 0, stream>>>(ps, psa, pt, pta, lab, pseudo,
                                                      facS, facT, epoch, per);
    combine_kernel<<<1, 64, 0, stream>>>(per, out);
}